// AvgSPP_14456859918372
// MI455X (gfx1250) — compile-verified
//
#include <hip/hip_runtime.h>

// ---------------------------------------------------------------------------
// AvgSPP for MI455X (gfx1250): 32x32 block-average pool + NN broadcast.
//
// Shapes: x[16,256,256,64] f32 -> out[16,256,256,64] f32, SCALE=8 => bin=32.
// Memory-bound: 256 MiB in + 256 MiB out => ~22 us floor @ 23.3 TB/s.
// Strategy: 1 workgroup per (b,u,v) bin (1024 blocks x 256 thr = 8 wave32).
//   - Reduction done with V_WMMA_F32_16X16X4_F32, A = ones, so
//     D[m][n] = sum_k B[k][n]: each WMMA sums 4 pixels x 16 channels, f32
//     exact. B/D share the same lane<->column striping, so "load your channel,
//     read your channel" is layout-permutation safe; all-ones A makes the
//     K-row assignment irrelevant.
//   - Wave wv: channel group (wv&3)*16..+15, row half (wv>>2)*16..+15.
//     128 chained WMMAs per wave cover 512 pixels.
//   - Partials combined via LDS (2 deterministic adds/channel), *1/1024,
//     then broadcast-written as coalesced float4 stores (512 B per wavefront store).
// ---------------------------------------------------------------------------

typedef __attribute__((ext_vector_type(2))) float v2f;
typedef __attribute__((ext_vector_type(4))) float v4f;
typedef __attribute__((ext_vector_type(8))) float v8f;

#define Bn 16
#define Hn 256
#define Wn 256
#define Cn 64
#define ROWSTRIDE (Wn * Cn)   // 16384 floats per image row

__global__ __launch_bounds__(256)
void avg_spp_kernel(const float* __restrict__ x, float* __restrict__ out) {
    __shared__ float pm[128];   // per-wave partial sums: pm[wv*16 + n]

    const int tid = threadIdx.x;
    const int bid = blockIdx.x;          // 0..1023 = b*64 + u*8 + v
    const int b   = bid >> 6;
    const int uv  = bid & 63;
    const int h0  = (uv >> 3) * 32;
    const int w0  = (uv & 7) * 32;

    const int wv   = tid >> 5;           // wave id 0..7
    const int lane = tid & 31;
    const int cg   = wv & 3;             // channel group (16 ch each)
    const int ph   = wv >> 2;            // row half (16 rows each)
    const int kh   = lane >> 4;          // which K-pair this lane feeds
    const int n    = lane & 15;          // column = channel within group
    const int ch   = cg * 16 + n;

    const size_t tile_base = (((size_t)b * Hn + h0) * Wn + w0) * Cn;

    v8f acc = {0.f, 0.f, 0.f, 0.f, 0.f, 0.f, 0.f, 0.f};
    const v2f ones = {1.0f, 1.0f};
    const int colOff = kh * 2 * Cn;      // lane half picks pixels {0,1} or {2,3}

    // Sum 512 pixels (16 rows x 32 cols) x 16 channels per wave.
    for (int r = 0; r < 16; ++r) {
        const float* rowptr =
            x + tile_base + (size_t)(ph * 16 + r) * ROWSTRIDE + ch + colOff;
        #pragma unroll
        for (int g = 0; g < 8; ++g) {        // 8 groups of 4 pixels per row
            v2f bmat;
            bmat.x = rowptr[g * 4 * Cn];      // pixel g*4 + 2*kh
            bmat.y = rowptr[g * 4 * Cn + Cn]; // pixel g*4 + 2*kh + 1
            // D = ones(16x4) * B(4x16) + C  => every D row = column sums
            acc = __builtin_amdgcn_wmma_f32_16x16x4_f32(
                false, ones, false, bmat, (short)0, acc, false, false);
        }
    }

    // D VGPR0, lanes 0..15 hold column n = lane: this wave's channel sums.
    if (lane < 16) pm[wv * 16 + n] = acc[0];
    __syncthreads();

    // Combine the two row-halves per channel and apply 1/1024 (block mean).
    if (tid < 64) pm[tid] = (pm[tid] + pm[tid + 64]) * (1.0f / 1024.0f);
    __syncthreads();

    // Broadcast: each thread owns 4 fixed channels, writes 64 float4 stores.
    const int c4 = (tid * 4) & 63;
    const v4f val = {pm[c4], pm[c4 + 1], pm[c4 + 2], pm[c4 + 3]};
    float* outp = out + tile_base + c4;
    #pragma unroll 4
    for (int iter = 0; iter < 64; ++iter) {
        const int p  = (tid >> 4) + iter * 16;  // pixel index in tile
        const int hl = p >> 5;
        const int wl = p & 31;
        *(v4f*)(outp + (size_t)hl * ROWSTRIDE + wl * Cn) = val;
    }
}

extern "C" void kernel_launch(void* const* d_in, const int* in_sizes, int n_in,
                              void* d_out, int out_size, void* d_ws, size_t ws_size,
                              hipStream_t stream) {
    (void)in_sizes; (void)n_in; (void)d_ws; (void)ws_size; (void)out_size;
    const float* x = (const float*)d_in[0];
    float* out = (float*)d_out;
    // 16 batches * 8 * 8 bins = 1024 workgroups, 256 threads (8 wave32) each.
    avg_spp_kernel<<<dim3(1024), dim3(256), 0, stream>>>(x, out);
}